// SPADEModel_58171037057165
// MI455X (gfx1250) — compile-verified
//
#include <hip/hip_runtime.h>
#include <hip/hip_bf16.h>

typedef __attribute__((ext_vector_type(2))) float v2f;
typedef __attribute__((ext_vector_type(8))) float v8f;

#define B_    4
#define C_    448
#define P_    3136
#define N_    200
#define K_IM  50
#define KTOT  (P_ * C_)          // 1,404,928
#define NTILES 13
#define NPAD   208               // 13 * 16
#define KSPLIT 128
#define KCHUNK (KTOT / KSPLIT)   // 10,976
#define KITER  (KCHUNK / 16)     // 686 iterations of K=16 (4 WMMAs each)

// ---- workspace layout (in floats) ----
#define ET_ELEMS   ((size_t)B_ * P_ * C_)      // e_t: 5,619,712
#define OFF_CROSS  ET_ELEMS                    // 4 * NPAD = 832
#define OFF_M2     (OFF_CROSS + 4 * NPAD)      // NPAD = 208
#define OFF_E2     (OFF_M2 + NPAD)             // 4
#define OFF_TOPK   (OFF_E2 + 4)                // 200 ints
#define ZERO_COUNT (4 * NPAD + NPAD + 4)       // 1044 floats to zero

__device__ __forceinline__ float wave_sum32(float v) {
    #pragma unroll
    for (int off = 16; off > 0; off >>= 1) v += __shfl_xor(v, off, 32);
    return v;
}

// ---------------- kernel 0: zero accumulators ----------------
__global__ void zero_kernel(float* __restrict__ p, int n) {
    int i = blockIdx.x * blockDim.x + threadIdx.x;
    if (i < n) p[i] = 0.0f;
}

// ---------------- kernel 1: [B,C,P] -> [B,P,C] transpose, also e2[b] ----------------
__global__ void transpose_kernel(const float* __restrict__ emb,
                                 float* __restrict__ e_t,
                                 float* __restrict__ e2) {
    __shared__ float tile[32][33];
    __shared__ float red[8];
    const int b  = blockIdx.z;
    const int c0 = blockIdx.y * 32;   // 448/32 = 14 tiles
    const int p0 = blockIdx.x * 32;   // 3136/32 = 98 tiles
    const int tx = threadIdx.x;       // 0..31
    const int ty = threadIdx.y;       // 0..7

    float ss = 0.0f;
    #pragma unroll
    for (int i = 0; i < 4; ++i) {
        int c = c0 + ty + i * 8;
        float v = emb[((size_t)b * C_ + c) * P_ + p0 + tx];  // coalesced over p
        tile[ty + i * 8][tx] = v;                            // tile[c_local][p_local]
        ss += v * v;
    }
    __syncthreads();
    #pragma unroll
    for (int i = 0; i < 4; ++i) {
        int p = p0 + ty + i * 8;
        e_t[((size_t)b * P_ + p) * C_ + c0 + tx] = tile[tx][ty + i * 8]; // coalesced over c
    }
    // block-reduce ss -> e2[b]
    int tid  = ty * 32 + tx;
    float ws_ = wave_sum32(ss);
    if ((tid & 31) == 0) red[tid >> 5] = ws_;
    __syncthreads();
    if (tid == 0) {
        float t = 0.0f;
        #pragma unroll
        for (int w = 0; w < 8; ++w) t += red[w];
        atomicAdd(&e2[b], t);
    }
}

// ---------------- kernel 2: split-K WMMA GEMM cross[4x200] + m2[200] ----------------
// No operand masking: A rows 4-15 duplicate rows 0-3 (their garbage lands in
// D rows we never read); B columns >= 200 read clamped row 0 (their garbage
// columns are excluded by the nvalid guard on the atomics). Data is finite,
// so no NaN/Inf contamination is possible.
__global__ void stage1_wmma_kernel(const float* __restrict__ mbank,
                                   const float* __restrict__ e_t,
                                   float* __restrict__ cross,
                                   float* __restrict__ m2) {
    const int lane   = threadIdx.x & 31;
    const int wave   = threadIdx.x >> 5;
    const int ntile  = blockIdx.x;                  // 0..12
    const int kchunk = blockIdx.y * 8 + wave;       // 0..127
    const size_t k0  = (size_t)kchunk * KCHUNK;

    const int m    = lane & 15;      // A row (batch, padded to 16)
    const int half = lane >> 4;      // selects K residue pair
    const int n    = ntile * 16 + (lane & 15);      // B column (memory image)
    const bool nvalid = (n < N_);

    const float* __restrict__ arow = e_t  + (size_t)(m & 3) * KTOT;          // clamp: rows 4-15 dup
    const float* __restrict__ brow = mbank + (size_t)(nvalid ? n : 0) * KTOT; // clamp: cols >=200 dup

    v8f acc = {};
    float m2p = 0.0f;

    for (int it = 0; it < KITER; ++it) {
        size_t base = k0 + (size_t)it * 16;
        #pragma unroll
        for (int j = 0; j < 4; ++j) {
            size_t ck = base + j * 4 + half * 2;
            v2f av = *(const v2f*)(arow + ck);   // 8B-aligned pair, feeds WMMA directly
            v2f bv = *(const v2f*)(brow + ck);
            m2p = fmaf(bv[0], bv[0], m2p);
            m2p = fmaf(bv[1], bv[1], m2p);
            acc = __builtin_amdgcn_wmma_f32_16x16x4_f32(
                      /*neg_a=*/false, av, /*neg_b=*/false, bv,
                      /*c_mod=*/(short)0, acc,
                      /*reuse_a=*/false, /*reuse_b=*/false);
        }
    }

    // m2: lanes L and L+16 hold partials for the same image
    m2p += __shfl_xor(m2p, 16, 32);
    if (half == 0 && nvalid) atomicAdd(&m2[n], m2p);

    // C/D layout: VGPR r, lanes 0-15 => D[r][col]; rows 0..3 are the batches
    if (half == 0 && nvalid) {
        #pragma unroll
        for (int r = 0; r < B_; ++r) atomicAdd(&cross[r * NPAD + n], acc[r]);
    }
}

// ---------------- kernel 3: image distances, top-50, pred_scores ----------------
__global__ void topk_kernel(const float* __restrict__ cross,
                            const float* __restrict__ m2,
                            const float* __restrict__ e2,
                            int* __restrict__ topk,
                            float* __restrict__ out) {
    int b = threadIdx.x;
    if (b >= B_) return;
    bool used[N_];
    for (int nn = 0; nn < N_; ++nn) used[nn] = false;
    const float e2b = e2[b];
    float s = 0.0f;
    for (int j = 0; j < K_IM; ++j) {
        float best = 3.402823466e38f;
        int   bi   = 0;
        for (int nn = 0; nn < N_; ++nn) {
            if (used[nn]) continue;
            float v = e2b + m2[nn] - 2.0f * cross[b * NPAD + nn];
            v = sqrtf(fmaxf(v, 0.0f));
            if (v < best) { best = v; bi = nn; }   // strict < : lowest index on ties
        }
        used[bi] = true;
        topk[b * K_IM + j] = bi;
        s += best;
    }
    out[b] = s / (float)K_IM;   // pred_scores[b]
}

// ---------------- kernel 4: patch scores (wave per (b,p)) ----------------
__global__ void stage2_patch_kernel(const float* __restrict__ mbank,
                                    const float* __restrict__ e_t,
                                    const int* __restrict__ topk,
                                    float* __restrict__ out) {
    const int lane = threadIdx.x & 31;
    const int wave = threadIdx.x >> 5;
    const int gp   = blockIdx.x * 8 + wave;   // 0 .. B*P-1 (exact: 1568*8 = 12544)
    const int b    = gp / P_;
    const int p    = gp % P_;

    // load e-vector for this (b,p): 14 floats per lane, coalesced
    float ev[14];
    const float* __restrict__ erow = e_t + ((size_t)b * P_ + p) * C_;
    float pe2p = 0.0f;
    #pragma unroll
    for (int i = 0; i < 14; ++i) {
        ev[i] = erow[lane + 32 * i];
        pe2p += ev[i] * ev[i];
    }
    const float pe2 = wave_sum32(pe2p);

    float mn = 3.402823466e38f;
    for (int k = 0; k < K_IM; ++k) {
        int img = topk[b * K_IM + k];
        const float* __restrict__ row = mbank + ((size_t)img * P_ + p) * C_;
        float acc = 0.0f;   // sum of (m^2 - 2*m*e)
        #pragma unroll
        for (int i = 0; i < 14; ++i) {
            float r = row[lane + 32 * i];
            acc = fmaf(r, r - 2.0f * ev[i], acc);
        }
        acc = wave_sum32(acc);
        float d = sqrtf(fmaxf(pe2 + acc, 0.0f));
        mn = fminf(mn, d);
    }
    if (lane == 0) out[4 + (size_t)b * P_ + p] = mn;   // patch_scores
}

extern "C" void kernel_launch(void* const* d_in, const int* in_sizes, int n_in,
                              void* d_out, int out_size, void* d_ws, size_t ws_size,
                              hipStream_t stream) {
    const float* emb   = (const float*)d_in[0];   // [4,448,56,56]
    const float* mbank = (const float*)d_in[1];   // [200,3136,448]
    float* ws    = (float*)d_ws;
    float* e_t   = ws;
    float* cross = ws + OFF_CROSS;
    float* m2    = ws + OFF_M2;
    float* e2    = ws + OFF_E2;
    int*   topk  = (int*)(ws + OFF_TOPK);
    float* out   = (float*)d_out;                 // [0..3]=pred, [4..]=patch

    zero_kernel<<<(ZERO_COUNT + 255) / 256, 256, 0, stream>>>(ws + OFF_CROSS, ZERO_COUNT);
    transpose_kernel<<<dim3(P_ / 32, C_ / 32, B_), dim3(32, 8), 0, stream>>>(emb, e_t, e2);
    stage1_wmma_kernel<<<dim3(NTILES, KSPLIT / 8), 256, 0, stream>>>(mbank, e_t, cross, m2);
    topk_kernel<<<1, 32, 0, stream>>>(cross, m2, e2, topk, out);
    stage2_patch_kernel<<<(B_ * P_) / 8, 256, 0, stream>>>(mbank, e_t, topk, out);
}